// MultiHeadAttention_7782480740629
// MI455X (gfx1250) — compile-verified
//
#include <hip/hip_runtime.h>
#include <cstddef>

// ---------------------------------------------------------------------------
// MultiHeadAttention pooling for MI455X (gfx1250, wave32).
// Memory-bound: ht = 512 MB f32, ~1 FLOP/byte -> single-pass flash-style
// online softmax so ht is streamed from HBM exactly once (~22us floor at
// 23.3 TB/s). Scores via V_WMMA_F32_16X16X4_F32 on LDS-staged 16x32 tiles.
// Total HBM traffic ~ 512 (ht) + 16 (p staged) + 16+16 (transpose) MB.
// ---------------------------------------------------------------------------

namespace {
constexpr int kB  = 16;
constexpr int kT  = 8192;
constexpr int kE  = 1024;
constexpr int kH  = 32;
constexpr int kHs = 32;
constexpr int kNW   = 8;    // waves per block (256 threads)
constexpr int kTPW  = 16;   // timesteps per WMMA tile
constexpr int kTSTR = 36;   // padded LDS tile row stride (floats): conflict-free, rows 16B aligned
constexpr float kScale = 0.17677669529663688f;  // 1/sqrt(d_k=32)
}

typedef float v2f __attribute__((ext_vector_type(2)));
typedef float v8f __attribute__((ext_vector_type(8)));

// Uniform broadcast from a fixed lane: v_readlane_b32 (VALU->SGPR),
// cheaper than ds_bpermute which burns LDS bandwidth.
__device__ __forceinline__ float rdlane(float v, int srclane) {
  return __uint_as_float(__builtin_amdgcn_readlane(__float_as_uint(v), srclane));
}

__global__ __launch_bounds__(256) void mha_pool_stage1(
    const float* __restrict__ ht, const float* __restrict__ query,
    float* __restrict__ out_ct, float* __restrict__ p_dst, int p_direct)
{
  __shared__ float s_scores[kT];                 // 32 KB raw scaled scores
  __shared__ float s_tile[kNW * kTPW * kTSTR];   // 18 KB per-wave v tiles
  __shared__ float s_q[kHs];
  __shared__ float s_m[kNW], s_s[kNW];
  __shared__ float s_ct[kNW * kHs];

  const int tid  = threadIdx.x;
  const int w    = tid >> 5;
  const int lane = tid & 31;
  const int b    = blockIdx.x >> 5;   // 512 blocks = (b,h)
  const int h    = blockIdx.x & 31;

  // query[d,h] at d*kH + h
  if (tid < kHs) s_q[tid] = query[tid * kH + h];
  __syncthreads();

  // B fragment (4x16, broadcast query chunk over all 16 N columns).
  // ISA layout: VGPR0 = K0 (lanes 0-15) | K2 (lanes 16-31); VGPR1 = K1 | K3.
  const int o = (lane >> 4) << 1;
  v2f bq[8];
#pragma unroll
  for (int k = 0; k < 8; ++k) {
    bq[k][0] = s_q[4 * k + o];
    bq[k][1] = s_q[4 * k + o + 1];
  }

  float* tw = &s_tile[w * kTPW * kTSTR];
  const float* htb = ht + (size_t)b * kT * kE + h * kHs;
  const int lrow = lane >> 3;         // 0..3
  const int lcol = (lane & 7) << 2;   // 0,4,...,28

  // Per-wave online softmax state (lane == head-dim d for ct_run).
  float m_run  = -__builtin_huge_valf();
  float s_run  = 0.f;
  float ct_run = 0.f;

  for (int c = w; c < kT / kTPW; c += kNW) {
    const int t0 = c * kTPW;

    // Stage 16x32 v-tile into padded LDS (coalesced 128B float4 rows).
#pragma unroll
    for (int i = 0; i < 4; ++i) {
      const int row = i * 4 + lrow;
      float4 v = *(const float4*)(htb + (size_t)(t0 + row) * kE + lcol);
      *(float4*)(tw + row * kTSTR + lcol) = v;
    }
    if (c + kNW < kT / kTPW)  // prefetch next chunk into near caches (uniform branch)
      __builtin_prefetch(htb + (size_t)(t0 + kNW * kTPW + lrow) * kE + lcol, 0, 3);
    // within-wave cross-lane LDS RAW: wait for tile stores before frag reads
    asm volatile("s_wait_dscnt 0" ::: "memory");

    // scores tile: D(16t x 16) = A(16t x 32d) * B(32d x 16) via 8x K=4 WMMA.
    v8f acc = {0.f, 0.f, 0.f, 0.f, 0.f, 0.f, 0.f, 0.f};
#pragma unroll
    for (int k = 0; k < 8; ++k) {
      const float* ap = tw + (lane & 15) * kTSTR + 4 * k + o;
      v2f a;
      a[0] = ap[0];
      a[1] = ap[1];
      acc = __builtin_amdgcn_wmma_f32_16x16x4_f32(false, a, false, bq[k],
                                                  (short)0, acc, false, false);
    }

    // Column N=0 of D: lane 0 holds M=0..7, lane 16 holds M=8..15.
    float sc[16];
#pragma unroll
    for (int r = 0; r < 8; ++r) {
      sc[r]     = rdlane(acc[r], 0)  * kScale;
      sc[r + 8] = rdlane(acc[r], 16) * kScale;
    }
    if (lane == 0) {
#pragma unroll
      for (int r = 0; r < 16; ++r) s_scores[t0 + r] = sc[r];
    }

    // Online (flash) update of max / sumexp / weighted-value accumulator.
    float cmax = sc[0];
#pragma unroll
    for (int r = 1; r < 16; ++r) cmax = fmaxf(cmax, sc[r]);
    const float m_new = fmaxf(m_run, cmax);
    const float f = __expf(m_run - m_new);   // first iter: exp(-inf)=0
    float psum = 0.f, cadd = 0.f;
#pragma unroll
    for (int r = 0; r < 16; ++r) {
      const float p = __expf(sc[r] - m_new);
      psum += p;
      cadd = fmaf(p, tw[r * kTSTR + lane], cadd);  // lane == d, conflict-free
    }
    s_run  = fmaf(s_run, f, psum);
    ct_run = fmaf(ct_run, f, cadd);
    m_run  = m_new;
  }

  // Cross-wave combine.
  if (lane == 0) { s_m[w] = m_run; s_s[w] = s_run; }
  s_ct[w * kHs + lane] = ct_run;
  __syncthreads();

  float M = s_m[0];
#pragma unroll
  for (int i = 1; i < kNW; ++i) M = fmaxf(M, s_m[i]);
  float S = 0.f;
#pragma unroll
  for (int i = 0; i < kNW; ++i) S += s_s[i] * __expf(s_m[i] - M);
  const float invS = 1.f / S;

  if (w == 0) {
    float csum = 0.f;
#pragma unroll
    for (int i = 0; i < kNW; ++i) csum += s_ct[i * kHs + lane] * __expf(s_m[i] - M);
    out_ct[b * (kH * kHs) + h * kHs + lane] = csum * invS;  // ct[b, h*32+d]
  }

  // p_attn. Staged path: coalesced [B][H][T] float4 writes into workspace
  // (transposed by stage 2). Direct fallback: strided [B][T][H] writes.
  if (p_direct) {
    float* dst = p_dst + (size_t)b * kT * kH + h;
    for (int t = tid; t < kT; t += 256)
      dst[(size_t)t * kH] = __expf(s_scores[t] - M) * invS;
  } else {
    float* dst = p_dst + (size_t)(b * kH + h) * kT;
    for (int t = tid * 4; t < kT; t += 1024) {
      float4 s4 = *(const float4*)&s_scores[t];
      float4 p4;
      p4.x = __expf(s4.x - M) * invS;
      p4.y = __expf(s4.y - M) * invS;
      p4.z = __expf(s4.z - M) * invS;
      p4.w = __expf(s4.w - M) * invS;
      *(float4*)&dst[t] = p4;
    }
  }
}

// ws[b][h][t]  ->  out_p[b][t][h]   (32h x 64t tiles through padded LDS)
__global__ __launch_bounds__(256) void mha_pool_stage2(
    const float* __restrict__ wsP, float* __restrict__ outP)
{
  __shared__ float ld[32 * 65];
  const int tid  = threadIdx.x;
  const int b    = blockIdx.x >> 7;     // kT/64 = 128 tiles per b
  const int t0   = (blockIdx.x & 127) * 64;

  // Read: float4 along t (512B/wave coalesced), scalar LDS stores (pad 65).
#pragma unroll
  for (int i = 0; i < 2; ++i) {
    const int idx  = tid + i * 256;     // 512 float4 per tile
    const int row  = idx >> 4;          // h
    const int col4 = (idx & 15) * 4;    // t
    float4 v = *(const float4*)&wsP[(size_t)(b * kH + row) * kT + t0 + col4];
    ld[row * 65 + col4 + 0] = v.x;
    ld[row * 65 + col4 + 1] = v.y;
    ld[row * 65 + col4 + 2] = v.z;
    ld[row * 65 + col4 + 3] = v.w;
  }
  __syncthreads();
  // Write: float4 along h (contiguous in out), strided LDS reads (conflict-free).
#pragma unroll
  for (int i = 0; i < 2; ++i) {
    const int idx  = tid + i * 256;
    const int rowq = (idx & 7) * 4;     // h base
    const int col  = idx >> 3;          // t local
    float4 v;
    v.x = ld[(rowq + 0) * 65 + col];
    v.y = ld[(rowq + 1) * 65 + col];
    v.z = ld[(rowq + 2) * 65 + col];
    v.w = ld[(rowq + 3) * 65 + col];
    *(float4*)&outP[((size_t)b * kT + t0 + col) * kH + rowq] = v;
  }
}

extern "C" void kernel_launch(void* const* d_in, const int* in_sizes, int n_in,
                              void* d_out, int out_size, void* d_ws, size_t ws_size,
                              hipStream_t stream) {
  (void)in_sizes; (void)n_in; (void)out_size;
  const float* ht    = (const float*)d_in[0];   // (16, 8192, 1024) f32
  const float* query = (const float*)d_in[1];   // (32, 32) f32

  float* out    = (float*)d_out;
  float* out_ct = out;                          // B*H*Hs = 16384 floats
  float* out_p  = out + kB * kH * kHs;          // B*T*H floats

  const size_t pbytes = (size_t)kB * kH * kT * sizeof(float);  // 16 MB
  const int p_direct = (ws_size < pbytes) ? 1 : 0;
  float* pdst = p_direct ? out_p : (float*)d_ws;

  mha_pool_stage1<<<dim3(kB * kH), dim3(256), 0, stream>>>(
      ht, query, out_ct, pdst, p_direct);
  if (!p_direct) {
    mha_pool_stage2<<<dim3(kB * (kT / 64)), dim3(256), 0, stream>>>(
        (const float*)d_ws, out_p);
  }
}